// ManifoldValuedVolterra_86973087744547
// MI455X (gfx1250) — compile-verified
//
#include <hip/hip_runtime.h>

typedef __attribute__((ext_vector_type(2))) float v2f;
typedef __attribute__((ext_vector_type(8))) float v8f;

namespace {

constexpr int Cc  = 4;
constexpr int Dd  = 64;
constexpr int Hh  = 64;
constexpr int Ww  = 64;
constexpr int Do  = 62;
constexpr int Ho  = 62;
constexpr int Wo  = 62;

// x[b][c][z][y][x][d] strides (floats)
constexpr long SX_y = 3L * Ww;
constexpr long SX_z = 3L * Ww * Hh;
constexpr long SX_c = 3L * Ww * Hh * Dd;
constexpr long SX_b = SX_c * Cc;

// out[b][o][z][y][x][d] strides (floats)
constexpr long SO_y = 3L * Wo;
constexpr long SO_z = 3L * Wo * Ho;
constexpr long SO_c = SO_z * Do;
constexpr long SO_b = SO_c * Cc;

template <int IMM>
__device__ __forceinline__ float swz(float x) {
  return __builtin_bit_cast(
      float, __builtin_amdgcn_ds_swizzle(__builtin_bit_cast(int, x), IMM));
}

constexpr int SWZ_SWAPX16 = 0x401F;  // xor 0x10, and 0x1f
constexpr int SWZ_QROT1   = 0x8039;  // quad perm (1,2,3,0)
constexpr int SWZ_QROT2   = 0x804E;  // quad perm (2,3,0,1)
constexpr int SWZ_QROT3   = 0x8093;  // quad perm (3,0,1,2)

} // namespace

__global__ __launch_bounds__(256)
void mv_volterra_wmma(const float* __restrict__ x,
                      const float* __restrict__ w1,
                      const float* __restrict__ w2,
                      const float* __restrict__ w3,
                      float* __restrict__ out)
{
  // stage the 3 weight tensors (C,K,K,K,C) = 432 floats each into LDS
  __shared__ float wlds[3][432];
  for (int t = threadIdx.x; t < 432; t += blockDim.x) {
    wlds[0][t] = w1[t];
    wlds[1][t] = w2[t];
    wlds[2][t] = w3[t];
  }
  __syncthreads();

  // block decode: bx = ((b*Do + zo)*Ho + yo)*2 + xblk
  int bx = blockIdx.x;
  const int xblk = bx & 1;
  int t2 = bx >> 1;
  const int yo = t2 % Ho;  t2 /= Ho;
  const int zo = t2 % Do;  t2 /= Do;
  const int b  = t2;

  const int wv    = threadIdx.x >> 5;   // wave in block (8 waves, 4 voxels each)
  const int lane  = threadIdx.x & 31;
  const int halfw = lane >> 4;          // 0: K=i{0,1}, 1: K=i{2,3}
  const int lid   = lane & 15;          // A-row M  and  B/D-column N
  const int v     = lid >> 2;           // voxel within wave (0..3)
  const int oc    = lid & 3;            // row channel o  == column component d
  const int i0    = halfw * 2;          // this lane's first input channel

  const int x0 = xblk * 32 + wv * 4;
  const int xo = x0 + v;
  const int xc = (xo < Wo) ? xo : (Wo - 1);   // clamp (stores masked later)

  const float* xb = x + (long)b * SX_b;

  // center point p[v][oc][:]
  const long pbase = (long)oc * SX_c + (long)(zo + 1) * SX_z +
                     (long)(yo + 1) * SX_y + (long)(xc + 1) * 3L;
  const float p0 = xb[pbase + 0];
  const float p1 = xb[pbase + 1];
  const float p2 = xb[pbase + 2];

  // constant identity-B for the  -s[o]*p[o,d]  correction WMMA:
  // B[k, (v,d)] = (k == d);  this lane supplies rows k = i0, i0+1, column d = oc
  v2f Bid;
  Bid.x = (oc == i0)     ? 1.0f : 0.0f;
  Bid.y = (oc == i0 + 1) ? 1.0f : 0.0f;

  v8f accv[3];
  #pragma unroll
  for (int s = 0; s < 3; ++s) accv[s] = (v8f){0, 0, 0, 0, 0, 0, 0, 0};

  const int widx0 = oc * 108 + i0;   // w[o, tap, i] = o*108 + tap*4 + i

  for (int tap = 0; tap < 27; ++tap) {
    const int kz = tap / 9;
    const int ky = (tap / 3) % 3;
    const int kx = tap % 3;
    const long qoff = (long)(zo + kz) * SX_z + (long)(yo + ky) * SX_y +
                      (long)(xc + kx) * 3L;
    const float* qa = xb + (long)i0 * SX_c + qoff;   // channel i0
    const float* qb = qa + SX_c;                     // channel i0+1
    const float qa0 = qa[0], qa1 = qa[1], qa2 = qa[2];
    const float qb0 = qb[0], qb1 = qb[1], qb2 = qb[2];

    // dots p_o . q_i  (this lane: (v, oc) x {i0, i0+1})
    float da = p0 * qa0 + p1 * qa1 + p2 * qa2;
    float db = p0 * qb0 + p1 * qb1 + p2 * qb2;
    da = fminf(fmaxf(da, -1.0f + 1e-6f), 1.0f - 1e-6f);
    db = fminf(fmaxf(db, -1.0f + 1e-6f), 1.0f - 1e-6f);
    // theta/sin(theta) = acos(dot) * rsqrt(1 - dot^2)
    const float ca = acosf(da) * rsqrtf(1.0f - da * da);
    const float cb = acosf(db) * rsqrtf(1.0f - db * db);

    // B1[k=i, (v,d)] = q[v][i][d]; this lane: column d = oc, rows i0, i0+1
    v2f Bq;
    Bq.x = (oc == 0) ? qa0 : (oc == 1) ? qa1 : (oc == 2) ? qa2 : 0.0f;
    Bq.y = (oc == 0) ? qb0 : (oc == 1) ? qb1 : (oc == 2) ? qb2 : 0.0f;

    // A2 rows carry -s * p[v][o][k]; lane supplies k = i0, i0+1 (k=3 -> 0)
    const float pk0 = (halfw == 0) ? p0 : p2;
    const float pk1 = (halfw == 0) ? p1 : 0.0f;

    const int wi = widx0 + tap * 4;
    #pragma unroll
    for (int s = 0; s < 3; ++s) {
      const float g0 = wlds[s][wi]     * ca;   // g = w * c
      const float g1 = wlds[s][wi + 1] * cb;
      // s[v,o] = sum_i g*dot : this half has i{0,1} (or {2,3}); add partner's
      float ps = g0 * da + g1 * db;
      ps += swz<SWZ_SWAPX16>(ps);

      v2f Ag; Ag.x = g0;        Ag.y = g1;         // A1[(v,o), i]
      v2f As; As.x = -ps * pk0; As.y = -ps * pk1;  // A2[(v,o), k]

      // acc += g x q     (D[(v,o),(v,d)] += sum_i g[v,o,i] q[v,i,d])
      accv[s] = __builtin_amdgcn_wmma_f32_16x16x4_f32(
          false, Ag, false, Bq, (short)0, accv[s], false, false);
      // acc += (-s p) x I  (D[(v,o),(v,d)] -= s[v,o] p[v,o,d])
      accv[s] = __builtin_amdgcn_wmma_f32_16x16x4_f32(
          false, As, false, Bid, (short)0, accv[s], false, false);
    }
  }

  // ---- extract the block-diagonal entries acc_s[v, o, d=oc] --------------
  // D layout: lanes 0-15 hold column N=lane, rows M=j in VGPR j;
  //           lanes 16-31 hold column N=lane-16, rows M=j+8.
  // Direct ext-vector element extraction (constant indices) keeps this in
  // registers — no memory round-trip.
  const bool useSwap = (v >= 2) != (halfw != 0);  // rows live in other half
  float r[3][4];
  #pragma unroll
  for (int s = 0; s < 3; ++s) {
    #pragma unroll
    for (int o = 0; o < 4; ++o) {       // row M = 4*v + o
      const float loA = accv[s][o];
      const float loB = accv[s][4 + o];
      const float swA = swz<SWZ_SWAPX16>(loA);
      const float swB = swz<SWZ_SWAPX16>(loB);
      const float loc = ((v & 1) == 0) ? loA : loB;
      const float swp = ((v & 1) == 0) ? swA : swB;
      r[s][o] = useSwap ? swp : loc;
    }
  }

  // v = log1 + log2*log3   (this lane holds component d = oc for 4 o's)
  float vt[4], sq[4];
  #pragma unroll
  for (int o = 0; o < 4; ++o) {
    float val = r[0][o] + r[1][o] * r[2][o];
    if (oc == 3) val = 0.0f;            // pad column carries garbage
    vt[o] = val;
    sq[o] = val * val;
  }

  // reload p components p[v][o][d=oc] for the exp map
  const long pb2 = (long)(zo + 1) * SX_z + (long)(yo + 1) * SX_y +
                   (long)(xc + 1) * 3L + oc;
  float pc[4];
  #pragma unroll
  for (int o = 0; o < 4; ++o) pc[o] = xb[(long)o * SX_c + pb2];

  // sphere exp + store; lanes 16-31 are duplicates, d=3 is pad, clamp edge
  if (halfw == 0 && oc < 3 && xo < Wo) {
    const long ob = (long)b * SO_b + (long)zo * SO_z + (long)yo * SO_y +
                    (long)xo * 3L + oc;
    #pragma unroll
    for (int o = 0; o < 4; ++o) {
      // ||v||^2 across the quad (lanes d=0..2 enabled; disabled lanes read 0)
      float nsq = sq[o] + swz<SWZ_QROT1>(sq[o]) + swz<SWZ_QROT2>(sq[o]) +
                  swz<SWZ_QROT3>(sq[o]);
      const float nrm = sqrtf(nsq + 1e-12f);
      const float cn  = cosf(nrm);
      const float sn  = sinf(nrm) / nrm;
      out[ob + (long)o * SO_c] = cn * pc[o] + sn * vt[o];
    }
  }
}

extern "C" void kernel_launch(void* const* d_in, const int* in_sizes, int n_in,
                              void* d_out, int out_size, void* d_ws, size_t ws_size,
                              hipStream_t stream) {
  (void)in_sizes; (void)n_in; (void)out_size; (void)d_ws; (void)ws_size;
  const float* x  = (const float*)d_in[0];
  const float* w1 = (const float*)d_in[1];
  const float* w2 = (const float*)d_in[2];
  const float* w3 = (const float*)d_in[3];
  float* out = (float*)d_out;

  // blocks: b(2) * zo(62) * yo(62) * x-halves(2); 256 thr = 8 waves * 4 voxels
  const int nblocks = 2 * Do * Ho * 2;
  hipLaunchKernelGGL(mv_volterra_wmma, dim3(nblocks), dim3(256), 0, stream,
                     x, w1, w2, w3, out);
}